// GCN_17162689314901
// MI455X (gfx1250) — compile-verified
//
#include <hip/hip_runtime.h>
#include <math.h>

#define NN   500000      // nodes
#define NE   16000000    // edges
#define NF   16          // input features
#define EMB  4
#define NG   1024        // graphs
#define HDIM 56          // pooled feature dim (EMB*14)
#define HPAD 64          // padded row stride for hcat / z

typedef __attribute__((ext_vector_type(2))) float v2f;
typedef __attribute__((ext_vector_type(8))) float v8f;

// ---- monotone uint encoding of float for atomicMax-based segment_max ----
__device__ __forceinline__ unsigned enc_f(float f) {
    unsigned u = __float_as_uint(f);
    return (u & 0x80000000u) ? ~u : (u | 0x80000000u);
}
__device__ __forceinline__ float dec_f(unsigned u) {
    return __uint_as_float((u & 0x80000000u) ? (u & 0x7FFFFFFFu) : ~u);
}
#define ENC_NEG_INF 0x007FFFFFu   // enc_f(-inf)

// =================== degree / norm precompute ===================
__global__ void k_init_deg(float* deg) {
    int n = blockIdx.x * blockDim.x + threadIdx.x;
    if (n < NN) deg[n] = 1.0f;                 // self-loop
}
__global__ void k_deg_accum(const int* __restrict__ dst, float* deg) {
    int e = blockIdx.x * blockDim.x + threadIdx.x;
    if (e < NE) atomicAdd(&deg[dst[e]], 1.0f);
}
__global__ void k_dis(const float* __restrict__ deg, float* dis) {
    int n = blockIdx.x * blockDim.x + threadIdx.x;
    if (n < NN) dis[n] = rsqrtf(deg[n]);       // deg >= 1 always
}
__global__ void k_norm(const int* __restrict__ src, const int* __restrict__ dst,
                       const float* __restrict__ dis, float* __restrict__ norm) {
    int e = blockIdx.x * blockDim.x + threadIdx.x;
    if (e < NE) norm[e] = dis[src[e]] * dis[dst[e]];
}

// =================== per-node linear  xw = h @ W  (WMMA f32 16x16x4) ===================
// One wave per 16-node tile. K in {16,4}, N = EMB(4) padded into the 16-wide tile.
template <int K>
__global__ void k_xw_wmma(const float* __restrict__ h, const float* __restrict__ W,
                          float* __restrict__ xw) {
    const int lane = threadIdx.x & 31;
    const int wid  = (blockIdx.x * blockDim.x + threadIdx.x) >> 5;
    const int NT   = NN / 16;                  // 31250 tiles, exact
    if (wid >= NT) return;                     // wave-uniform guard
    const int half = lane >> 4;                // 0: K lo pair, 1: K hi pair
    const int li   = lane & 15;
    const int n    = li;                       // B/D column
    const int nc   = (n < EMB) ? n : 0;        // clamped (in-bounds) column
    const int m0   = wid * 16;

    v8f acc = {};
#pragma unroll
    for (int kc = 0; kc < K / 4; ++kc) {
        const int kb = kc * 4 + half * 2;
        const int m  = m0 + li;
        v2f a, b;
        a.x = h[m * K + kb];
        a.y = h[m * K + kb + 1];
        float bx = W[kb * EMB + nc];
        float by = W[(kb + 1) * EMB + nc];
        b.x = (n < EMB) ? bx : 0.0f;
        b.y = (n < EMB) ? by : 0.0f;
        acc = __builtin_amdgcn_wmma_f32_16x16x4_f32(false, a, false, b,
                                                    (short)0, acc, false, false);
    }
    if (n < EMB) {
#pragma unroll
        for (int v = 0; v < 8; ++v) {
            const int m = m0 + v + half * 8;
            xw[m * EMB + n] = acc[v];
        }
    }
}

// h_out[n] = b + (1/deg[n]) * xw[n]     (bias + self-loop term, dis^2 = 1/deg)
__global__ void k_init_hout(const float* __restrict__ xw, const float* __restrict__ dis,
                            const float* __restrict__ b, float* __restrict__ hout) {
    int n = blockIdx.x * blockDim.x + threadIdx.x;
    if (n >= NN) return;
    float sn = dis[n] * dis[n];
#pragma unroll
    for (int j = 0; j < EMB; ++j)
        hout[n * EMB + j] = b[j] + sn * xw[n * EMB + j];
}

// edge scatter: h_out[dst] += norm * xw[src]
__global__ void k_scatter(const int* __restrict__ src, const int* __restrict__ dst,
                          const float* __restrict__ norm, const float4* __restrict__ xw4,
                          float* __restrict__ hout) {
    int e = blockIdx.x * blockDim.x + threadIdx.x;
    if (e >= NE) return;
    int s = src[e], d = dst[e];
    float w = norm[e];
    float4 v = xw4[s];
    atomicAdd(&hout[d * 4 + 0], w * v.x);
    atomicAdd(&hout[d * 4 + 1], w * v.y);
    atomicAdd(&hout[d * 4 + 2], w * v.z);
    atomicAdd(&hout[d * 4 + 3], w * v.w);
}

// =================== pooling ===================
__global__ void k_init_hcat(unsigned* __restrict__ hcat_u, float* __restrict__ counts) {
    int i = blockIdx.x * blockDim.x + threadIdx.x;
    if (i < NG * HPAD) {
        int c = i & (HPAD - 1);
        bool ismax = (c < HDIM) && ((c & 7) < 4);
        hcat_u[i] = ismax ? ENC_NEG_INF : 0u;   // 0u == bits of 0.0f for mean/pad cols
    }
    if (i < NG) counts[i] = 0.0f;
}
__global__ void k_counts(const int* __restrict__ batch, float* counts) {
    int n = blockIdx.x * blockDim.x + threadIdx.x;
    if (n < NN) atomicAdd(&counts[batch[n]], 1.0f);
}
// slot = conv index (0..6): max -> cols [8s,8s+4), sum -> cols [8s+4,8s+8)
__global__ void k_pool(const float4* __restrict__ h4, const int* __restrict__ batch,
                       unsigned* __restrict__ hcat_u, float* __restrict__ hcat_f, int slot) {
    int n = blockIdx.x * blockDim.x + threadIdx.x;
    if (n >= NN) return;
    int g = batch[n];
    float4 v = h4[n];
    float vv[4] = {v.x, v.y, v.z, v.w};
    unsigned* pm = hcat_u + g * HPAD + slot * 8;
    float*    ps = hcat_f + g * HPAD + slot * 8 + 4;
#pragma unroll
    for (int j = 0; j < 4; ++j) {
        atomicMax(&pm[j], enc_f(vv[j]));
        atomicAdd(&ps[j], vv[j]);
    }
}
__global__ void k_finalize_hcat(unsigned* __restrict__ hcat_u, float* __restrict__ hcat_f,
                                const float* __restrict__ counts) {
    int i = blockIdx.x * blockDim.x + threadIdx.x;
    if (i >= NG * HPAD) return;
    int c = i & (HPAD - 1), g = i >> 6;
    if (c >= HDIM) return;                      // pad cols stay 0
    if ((c & 7) < 4) hcat_f[i] = dec_f(hcat_u[i]);          // segment_max
    else             hcat_f[i] = hcat_f[i] / fmaxf(counts[g], 1.0f);  // mean
}

// =================== MLP head: z = relu(hcat @ Wout + bout)  (WMMA) ===================
// 64 M-tiles x 4 N-tiles = 256 waves; K = 56 -> 14 unrolled WMMA per wave.
__global__ void k_mlp_wmma(const float* __restrict__ hcat, const float* __restrict__ Wout,
                           const float* __restrict__ bout, float* __restrict__ z) {
    const int lane  = threadIdx.x & 31;
    const int wid   = (blockIdx.x * blockDim.x + threadIdx.x) >> 5;  // 0..255 exact
    const int mtile = wid >> 2;
    const int ntile = wid & 3;
    const int half  = lane >> 4;
    const int li    = lane & 15;
    const int n     = ntile * 16 + li;
    const bool nv   = (n < HDIM);
    const int nc    = nv ? n : 0;

    v8f acc = {};
#pragma unroll
    for (int kc = 0; kc < HDIM / 4; ++kc) {     // 14 chunks
        const int kb = kc * 4 + half * 2;
        const int m  = mtile * 16 + li;
        v2f a, b;
        a.x = hcat[m * HPAD + kb];
        a.y = hcat[m * HPAD + kb + 1];
        float bx = Wout[kb * HDIM + nc];
        float by = Wout[(kb + 1) * HDIM + nc];
        b.x = nv ? bx : 0.0f;
        b.y = nv ? by : 0.0f;
        acc = __builtin_amdgcn_wmma_f32_16x16x4_f32(false, a, false, b,
                                                    (short)0, acc, false, false);
    }
    if (nv) {
        float bo = bout[n];
#pragma unroll
        for (int v = 0; v < 8; ++v) {
            int m = mtile * 16 + v + half * 8;
            z[m * HPAD + n] = fmaxf(acc[v] + bo, 0.0f);
        }
    }
}

// logit = z @ Wout2 + bout2 ; out = [sigmoid(logit) | logit]
__global__ void k_final(const float* __restrict__ z, const float* __restrict__ Wout2,
                        const float* __restrict__ bout2, float* __restrict__ out) {
    int g = blockIdx.x * blockDim.x + threadIdx.x;
    if (g >= NG) return;
    float acc = bout2[0];
#pragma unroll 8
    for (int k = 0; k < HDIM; ++k) acc += z[g * HPAD + k] * Wout2[k];
    out[g]      = 1.0f / (1.0f + __expf(-acc));
    out[NG + g] = acc;
}

// =================== host launch ===================
static inline int cdiv(int a, int b) { return (a + b - 1) / b; }

extern "C" void kernel_launch(void* const* d_in, const int* in_sizes, int n_in,
                              void* d_out, int out_size, void* d_ws, size_t ws_size,
                              hipStream_t stream) {
    const float* x     = (const float*)d_in[0];
    const int*   eidx  = (const int*)  d_in[1];
    const int*   batch = (const int*)  d_in[2];
    /* d_in[3] = num_graphs scalar (compile-time 1024) */
    const float* W0    = (const float*)d_in[4];
    const float* b0    = (const float*)d_in[5];
    const float* Wc    = (const float*)d_in[6];   // [6,4,4]
    const float* bc    = (const float*)d_in[7];   // [6,4]
    const float* Wout  = (const float*)d_in[8];   // [56,56]
    const float* bout  = (const float*)d_in[9];
    const float* Wout2 = (const float*)d_in[10];  // [56,1]
    const float* bout2 = (const float*)d_in[11];

    const int* src = eidx;
    const int* dst = eidx + NE;

    // workspace carve-up (all region sizes are multiples of 4 floats -> 16B aligned)
    float* ws     = (float*)d_ws;
    float* deg    = ws;  ws += NN;
    float* dis    = ws;  ws += NN;
    float* norm   = ws;  ws += NE;
    float* xw     = ws;  ws += NN * EMB;
    float* hA     = ws;  ws += NN * EMB;
    float* hB     = ws;  ws += NN * EMB;
    float* hcat   = ws;  ws += NG * HPAD;
    float* counts = ws;  ws += NG;
    float* zbuf   = ws;  ws += NG * HPAD;

    const int BT  = 256;
    const int XWB = cdiv(NN / 16, BT / 32);     // wmma blocks: 8 waves/block

    // degrees, norms, pooling init
    k_init_deg  <<<cdiv(NN, BT), BT, 0, stream>>>(deg);
    k_deg_accum <<<cdiv(NE, BT), BT, 0, stream>>>(dst, deg);
    k_dis       <<<cdiv(NN, BT), BT, 0, stream>>>(deg, dis);
    k_norm      <<<cdiv(NE, BT), BT, 0, stream>>>(src, dst, dis, norm);
    k_init_hcat <<<cdiv(NG * HPAD, BT), BT, 0, stream>>>((unsigned*)hcat, counts);
    k_counts    <<<cdiv(NN, BT), BT, 0, stream>>>(batch, counts);

    // conv0: x[NN,16] -> hA[NN,4]
    k_xw_wmma<NF><<<XWB, BT, 0, stream>>>(x, W0, xw);
    k_init_hout  <<<cdiv(NN, BT), BT, 0, stream>>>(xw, dis, b0, hA);
    k_scatter    <<<cdiv(NE, BT), BT, 0, stream>>>(src, dst, norm, (const float4*)xw, hA);
    k_pool       <<<cdiv(NN, BT), BT, 0, stream>>>((const float4*)hA, batch,
                                                   (unsigned*)hcat, hcat, 0);
    // convs 1..6 (ping-pong hA/hB)
    float* hin = hA; float* hout = hB;
    for (int i = 0; i < 6; ++i) {
        k_xw_wmma<EMB><<<XWB, BT, 0, stream>>>(hin, Wc + i * EMB * EMB, xw);
        k_init_hout   <<<cdiv(NN, BT), BT, 0, stream>>>(xw, dis, bc + i * EMB, hout);
        k_scatter     <<<cdiv(NE, BT), BT, 0, stream>>>(src, dst, norm,
                                                        (const float4*)xw, hout);
        k_pool        <<<cdiv(NN, BT), BT, 0, stream>>>((const float4*)hout, batch,
                                                        (unsigned*)hcat, hcat, i + 1);
        float* t = hin; hin = hout; hout = t;
    }

    // head
    k_finalize_hcat<<<cdiv(NG * HPAD, BT), BT, 0, stream>>>((unsigned*)hcat, hcat, counts);
    k_mlp_wmma     <<<(NG / 16) * 4 * 32 / BT, BT, 0, stream>>>(hcat, Wout, bout, zbuf);
    k_final        <<<cdiv(NG, BT), BT, 0, stream>>>(zbuf, Wout2, bout2, (float*)d_out);
}